// TabularEncoder_3659312136363
// MI455X (gfx1250) — compile-verified
//
#include <hip/hip_runtime.h>
#include <hip/hip_bf16.h>

typedef __attribute__((ext_vector_type(2))) float v2f;
typedef __attribute__((ext_vector_type(8))) float v8f;

#define EMB 128
#define HID 11
#define TOK_PER_TILE 16
#define WAVES_PER_BLOCK 8

// One wave32 processes a tile of 16 tokens:
//   h[16 x 12]  (tanh MLP, K padded 11->12)
//   C[16 x 128] = h @ W2  via 8 N-tiles x 3 chained V_WMMA_F32_16X16X4_F32
//   out = C * category_mask + emb_table[var_id]
__global__ __launch_bounds__(256) void cve_wmma_kernel(
    const float* __restrict__ value,
    const int*   __restrict__ var_id,
    const int*   __restrict__ cat_mask,
    const float* __restrict__ W1,
    const float* __restrict__ b1,
    const float* __restrict__ W2,
    const float* __restrict__ emb,
    float* __restrict__ out,
    float* __restrict__ mask_out,
    int n_tok)
{
    const int lane = threadIdx.x & 31;
    const int wave = threadIdx.x >> 5;
    const int tile = blockIdx.x * WAVES_PER_BLOCK + wave;
    const int base = tile * TOK_PER_TILE;
    if (base >= n_tok) return;                 // uniform per wave

    const int hi  = lane >> 4;                 // half-wave select (0/1)
    const int col = lane & 15;                 // N / M-row index within 16

    // ---------- A matrix: h = tanh(v*W1 + b1), rows = tokens ----------
    // 32-bit A 16x4 layout: lane holds row M=col; VGPR j holds K = 2*hi + j.
    // Chunk kc covers global K = 4*kc .. 4*kc+3.
    const float v = value[base + col];
    v2f a[3];
#pragma unroll
    for (int kc = 0; kc < 3; ++kc) {
        const int k0 = kc * 4 + hi * 2;
        const int k1 = k0 + 1;
        const int k0c = (k0 < HID) ? k0 : 0;   // clamp to avoid OOB
        const int k1c = (k1 < HID) ? k1 : 0;
        const float h0 = tanhf(v * W1[k0c] + b1[k0c]);
        const float h1 = tanhf(v * W1[k1c] + b1[k1c]);
        a[kc].x = (k0 < HID) ? h0 : 0.0f;      // zero-pad K=11
        a[kc].y = (k1 < HID) ? h1 : 0.0f;
    }

    // ---------- per-row (C-layout) token data: row m = i + 8*hi ----------
    float cm[8];
    int   vid[8];
#pragma unroll
    for (int i = 0; i < 8; ++i) {
        const int t = base + i + hi * 8;
        cm[i]  = (float)cat_mask[t];
        vid[i] = var_id[t];
    }

    // padding_mask: clip(var_id, 0, 1) -> one write per token (lanes 0-15)
    if (hi == 0) {
        const int t = base + col;
        mask_out[t] = (var_id[t] > 0) ? 1.0f : 0.0f;
    }

    // ---------- 8 N-tiles of 16 emb dims each ----------
#pragma unroll
    for (int nt = 0; nt < 8; ++nt) {
        v8f c = {};
#pragma unroll
        for (int kc = 0; kc < 3; ++kc) {
            // B 4x16 layout: VGPR j holds B[K = 2*hi + j][N = col]
            const int k0 = kc * 4 + hi * 2;
            const int k1 = k0 + 1;
            const int k0c = (k0 < HID) ? k0 : 0;
            const int k1c = (k1 < HID) ? k1 : 0;
            const float w0 = W2[k0c * EMB + nt * 16 + col];
            const float w1 = W2[k1c * EMB + nt * 16 + col];
            v2f b;
            b.x = (k0 < HID) ? w0 : 0.0f;
            b.y = (k1 < HID) ? w1 : 0.0f;
            // D = A x B + C   (16x16x4 f32)
            c = __builtin_amdgcn_wmma_f32_16x16x4_f32(
                    /*neg_a=*/false, a[kc],
                    /*neg_b=*/false, b,
                    /*c_mod=*/(short)0, c,
                    /*reuse_a=*/false, /*reuse_b=*/false);
        }
        // combine with mask + embedding gather, store 16x16 tile
#pragma unroll
        for (int i = 0; i < 8; ++i) {
            const int row = i + hi * 8;
            const int t   = base + row;
            const float e = emb[(size_t)vid[i] * EMB + nt * 16 + col];
            out[(size_t)t * EMB + nt * 16 + col] = c[i] * cm[i] + e;
        }
    }
}

extern "C" void kernel_launch(void* const* d_in, const int* in_sizes, int n_in,
                              void* d_out, int out_size, void* d_ws, size_t ws_size,
                              hipStream_t stream) {
    const float* value    = (const float*)d_in[0];   // [4096,200]
    const int*   var_id   = (const int*)  d_in[1];   // [4096,200]
    const int*   cat_mask = (const int*)  d_in[2];   // [4096,200]
    const float* W1       = (const float*)d_in[3];   // [1,11]
    const float* b1       = (const float*)d_in[4];   // [11]
    const float* W2       = (const float*)d_in[5];   // [11,128]
    const float* emb      = (const float*)d_in[6];   // [1001,128]

    const int n_tok = in_sizes[0];                   // 819200
    float* out      = (float*)d_out;                 // sum_emb [n_tok,128]
    float* mask_out = out + (size_t)n_tok * EMB;     // padding_mask [n_tok]

    const int n_tiles = (n_tok + TOK_PER_TILE - 1) / TOK_PER_TILE;
    const int blocks  = (n_tiles + WAVES_PER_BLOCK - 1) / WAVES_PER_BLOCK;

    cve_wmma_kernel<<<blocks, 256, 0, stream>>>(
        value, var_id, cat_mask, W1, b1, W2, emb, out, mask_out, n_tok);
}